// Loss_54494545052046
// MI455X (gfx1250) — compile-verified
//
#include <hip/hip_runtime.h>
#include <math.h>

// ---------------------------------------------------------------------------
// Loss kernel for MI455X (gfx1250, wave32).
// recon = mean(|pm - tm|) + 0.1 * mean((pd - td)^2)   over (64,4096,150) f32.
// Memory-bound: 314.6 MB read @ 23.3 TB/s -> ~13.5us floor. Single streaming
// pass, LDS-staged tiles, deterministic tree reduction; final 256-way sum done
// with chained V_WMMA_F32_16X16X4_F32 (exact f32 MAC).
// ---------------------------------------------------------------------------

#define FEAT        150
#define NUM_BONES   50
#define ROWS_PER_BLK 32
#define TILE_FLOATS (ROWS_PER_BLK * FEAT)      // 4800 floats = 19200 B (16B aligned)
#define TILE_VEC4   (TILE_FLOATS / 4)          // 1200 float4
#define THREADS     256
#define F32_TINY    1.17549435e-38f

typedef float v2f __attribute__((ext_vector_type(2)));
typedef float v8f __attribute__((ext_vector_type(8)));

// 50-bone skeleton (body + two 21-bone hands + pad bone (0,2))
__constant__ int c_BA[NUM_BONES] = {
    0,1,2,3,1,5,6,
    7, 8,9,10,11, 8,13,14,15, 8,17,18,19, 8,21,22,23, 8,25,26,27,
    4, 29,30,31,32, 29,34,35,36, 29,38,39,40, 29,42,43,44, 29,46,47,48,
    0
};
__constant__ int c_BB[NUM_BONES] = {
    1,2,3,4,5,6,7,
    8, 9,10,11,12, 13,14,15,16, 17,18,19,20, 21,22,23,24, 25,26,27,28,
    29, 30,31,32,33, 34,35,36,37, 38,39,40,41, 42,43,44,45, 46,47,48,49,
    2
};

// ---------------------------------------------------------------------------
// Kernel 1: per-block partial sums (L1 term, squared-direction term).
// Each block: 32 rows of 150 floats, staged in LDS.
// ---------------------------------------------------------------------------
__global__ __launch_bounds__(THREADS)
void loss_partial_kernel(const float* __restrict__ preds,
                         const float* __restrict__ targets,
                         float2* __restrict__ partials)
{
    __shared__ float4 spm4[TILE_VEC4];   // masked preds tile
    __shared__ float4 stm4[TILE_VEC4];   // raw targets tile (== tm, mask is identity on targets)
    __shared__ float2 sred[THREADS];

    const int tid  = threadIdx.x;
    const size_t tileBase = (size_t)blockIdx.x * TILE_FLOATS;

    const float4* __restrict__ p4 = (const float4*)(preds   + tileBase);
    const float4* __restrict__ t4 = (const float4*)(targets + tileBase);

    float acc1 = 0.0f;   // sum |pm - tm|
    float acc2 = 0.0f;   // sum ((pd - td)^2)

    // Phase 1: coalesced b128 streaming load -> LDS, fused L1 accumulation.
    for (int i = tid; i < TILE_VEC4; i += THREADS) {
        float4 tv = t4[i];
        float4 pv = p4[i];
        float4 pm;
        pm.x = (tv.x != 0.0f) ? pv.x : 0.0f;
        pm.y = (tv.y != 0.0f) ? pv.y : 0.0f;
        pm.z = (tv.z != 0.0f) ? pv.z : 0.0f;
        pm.w = (tv.w != 0.0f) ? pv.w : 0.0f;
        spm4[i] = pm;
        stm4[i] = tv;
        acc1 += fabsf(pm.x - tv.x) + fabsf(pm.y - tv.y)
              + fabsf(pm.z - tv.z) + fabsf(pm.w - tv.w);
    }
    __syncthreads();

    const float* __restrict__ spm = (const float*)spm4;
    const float* __restrict__ stm = (const float*)stm4;

    // Phase 2: 32 rows x 50 bones = 1600 bone tasks.
    for (int t = tid; t < ROWS_PER_BLK * NUM_BONES; t += THREADS) {
        const int row  = t / NUM_BONES;
        const int bone = t - row * NUM_BONES;
        const int rb   = row * FEAT;
        const int ja   = rb + 3 * c_BA[bone];
        const int jb   = rb + 3 * c_BB[bone];
        const int mb   = rb + 3 * bone;          // mask indexed by (bone, comp)

        float dp0 = spm[ja + 0] - spm[jb + 0];
        float dp1 = spm[ja + 1] - spm[jb + 1];
        float dp2 = spm[ja + 2] - spm[jb + 2];
        float dt0 = stm[ja + 0] - stm[jb + 0];
        float dt1 = stm[ja + 1] - stm[jb + 1];
        float dt2 = stm[ja + 2] - stm[jb + 2];

        float lp = sqrtf(dp0 * dp0 + dp1 * dp1 + dp2 * dp2) + F32_TINY;
        float lt = sqrtf(dt0 * dt0 + dt1 * dt1 + dt2 * dt2) + F32_TINY;

        float m0 = (stm[mb + 0] != 0.0f) ? 1.0f : 0.0f;
        float m1 = (stm[mb + 1] != 0.0f) ? 1.0f : 0.0f;
        float m2 = (stm[mb + 2] != 0.0f) ? 1.0f : 0.0f;

        float e0 = (dp0 / lp - dt0 / lt) * m0;
        float e1 = (dp1 / lp - dt1 / lt) * m1;
        float e2 = (dp2 / lp - dt2 / lt) * m2;
        acc2 += e0 * e0 + e1 * e1 + e2 * e2;
    }

    // Block tree reduction (deterministic).
    sred[tid] = make_float2(acc1, acc2);
    __syncthreads();
    for (int s = THREADS / 2; s > 0; s >>= 1) {
        if (tid < s) {
            sred[tid].x += sred[tid + s].x;
            sred[tid].y += sred[tid + s].y;
        }
        __syncthreads();
    }
    if (tid == 0) partials[blockIdx.x] = sred[0];
}

// ---------------------------------------------------------------------------
// WMMA-based exact f32 sum of 256 shared-memory values (wave 0 only).
// A = ones(16x4)  =>  D[i][j] = colsum_j(B), so  Sum(all D regs, all lanes)
// = 16 * Sum(all 64 B entries) independent of the VGPR<->matrix layout.
// Chain 4 WMMAs over C to cover 256 values, then shfl-reduce, divide by 16.
// ---------------------------------------------------------------------------
__device__ inline float wmma_sum256(const float* __restrict__ s, int lane)
{
    v2f a; a.x = 1.0f; a.y = 1.0f;
    v8f c = {};
    #pragma unroll
    for (int step = 0; step < 4; ++step) {
        v2f b;
        b.x = s[step * 64 + lane];
        b.y = s[step * 64 + 32 + lane];
        c = __builtin_amdgcn_wmma_f32_16x16x4_f32(
                false, a, false, b, (short)0, c, false, false);
    }
    float loc = c[0] + c[1] + c[2] + c[3] + c[4] + c[5] + c[6] + c[7];
    #pragma unroll
    for (int m = 16; m >= 1; m >>= 1)
        loc += __shfl_xor(loc, m, 32);
    return loc * 0.0625f;   // /16, exact
}

// ---------------------------------------------------------------------------
// Kernel 2: reduce per-block partials to the scalar loss.
// ---------------------------------------------------------------------------
__global__ __launch_bounds__(THREADS)
void loss_final_kernel(const float2* __restrict__ partials,
                       int nblocks,
                       float inv_n,
                       float* __restrict__ out)
{
    __shared__ float s1[THREADS];
    __shared__ float s2[THREADS];

    const int tid = threadIdx.x;
    float a1 = 0.0f, a2 = 0.0f;
    for (int i = tid; i < nblocks; i += THREADS) {
        float2 v = partials[i];
        a1 += v.x;
        a2 += v.y;
    }
    s1[tid] = a1;
    s2[tid] = a2;
    __syncthreads();

    if (tid < 32) {   // wave 0: EXEC all ones (wave32), WMMA-legal
        float total1 = wmma_sum256(s1, tid);
        float total2 = wmma_sum256(s2, tid);
        if (tid == 0) {
            double r = (double)total1 * (double)inv_n
                     + 0.1 * (double)total2 * (double)inv_n;
            out[0] = (float)r;
        }
    }
}

// ---------------------------------------------------------------------------
extern "C" void kernel_launch(void* const* d_in, const int* in_sizes, int n_in,
                              void* d_out, int out_size, void* d_ws, size_t ws_size,
                              hipStream_t stream)
{
    const float* preds   = (const float*)d_in[0];
    const float* targets = (const float*)d_in[1];
    float*       out     = (float*)d_out;
    float2*      part    = (float2*)d_ws;

    const int n      = in_sizes[0];            // 39,321,600
    const int rows   = n / FEAT;               // 262,144
    const int blocks = rows / ROWS_PER_BLK;    // 8,192

    loss_partial_kernel<<<blocks, THREADS, 0, stream>>>(preds, targets, part);
    loss_final_kernel<<<1, THREADS, 0, stream>>>(part, blocks, 1.0f / (float)n, out);
}